// dgl_SAGE_10282151707715
// MI455X (gfx1250) — compile-verified
//
#include <hip/hip_runtime.h>
#include <hip/hip_bf16.h>

typedef float v2f __attribute__((ext_vector_type(2)));
typedef float v8f __attribute__((ext_vector_type(8)));

#define NN 50000
#define NE 800000
#define FD 128
#define LDSROW 132   // 128 + 4 pad floats to break LDS bank conflicts

__device__ __forceinline__ void atomadd_f32(float* p, float v) {
    // relaxed, agent scope -> global_atomic_add_f32 (no CAS loop)
    __hip_atomic_fetch_add(p, v, __ATOMIC_RELAXED, __HIP_MEMORY_SCOPE_AGENT);
}

// deg[i] = 2.0  (self-loop counted in numerator and in the reference's "+1" divisor)
__global__ void k_init_deg(float* __restrict__ deg) {
    int i = blockIdx.x * blockDim.x + threadIdx.x;
    if (i < NN) deg[i] = 2.0f;
}

// deg[dst[e]] += 1 per edge
__global__ void k_count_deg(const int* __restrict__ dst, float* __restrict__ deg) {
    int e = blockIdx.x * blockDim.x + threadIdx.x;
    if (e < NE) atomadd_f32(&deg[dst[e]], 1.0f);
}

// neigh = 2 * x  (vectorized float4)
__global__ void k_init_neigh(const float* __restrict__ x, float* __restrict__ neigh) {
    int i = blockIdx.x * blockDim.x + threadIdx.x;   // indexes float4
    const int total4 = NN * FD / 4;
    if (i < total4) {
        float4 v = ((const float4*)x)[i];
        float4 o = { 2.0f * v.x, 2.0f * v.y, 2.0f * v.z, 2.0f * v.w };
        ((float4*)neigh)[i] = o;
    }
}

// one wave per edge: lane l moves features [4l .. 4l+3]
__global__ void k_aggregate(const float* __restrict__ x, const int* __restrict__ src,
                            const int* __restrict__ dst, float* __restrict__ neigh) {
    int gid  = blockIdx.x * blockDim.x + threadIdx.x;
    int e    = gid >> 5;
    int lane = gid & 31;
    if (e >= NE) return;
    int s = src[e];
    int d = dst[e];
    float4 v = ((const float4*)(x + (size_t)s * FD))[lane];
    float* p = neigh + (size_t)d * FD + lane * 4;
    atomadd_f32(p + 0, v.x);
    atomadd_f32(p + 1, v.y);
    atomadd_f32(p + 2, v.z);
    atomadd_f32(p + 3, v.w);
}

// out[m][n] = act( sum_k (neigh[m][k]/deg[m]) * W[n][k] + bias[n] )
// grid.x = 3125 M-tiles; block = (32, NOUT/16) waves; K = FD = 128.
// Normalization is fused into the LDS staging of the 16x128 A-tile.
template <int NOUT, bool RELU>
__global__ void k_gemm_wmma(const float* __restrict__ neigh, const float* __restrict__ deg,
                            const float* __restrict__ W, const float* __restrict__ bias,
                            float* __restrict__ out) {
    __shared__ float aTile[16 * LDSROW];

    const int m0   = blockIdx.x * 16;
    const int tid  = threadIdx.y * 32 + threadIdx.x;
    const int nthr = (NOUT / 16) * 32;

    // stage 16 rows x 128 cols of h = neigh/deg into LDS as float4 (32 float4 per row)
    for (int idx = tid; idx < 16 * (FD / 4); idx += nthr) {
        int row = idx >> 5;
        int c4  = idx & 31;
        float  d = deg[m0 + row];
        float4 v = ((const float4*)(neigh + (size_t)(m0 + row) * FD))[c4];
        float4 o = { v.x / d, v.y / d, v.z / d, v.w / d };
        *(float4*)&aTile[row * LDSROW + c4 * 4] = o;
    }
    __syncthreads();

    const int lane = threadIdx.x;     // wave32: threadIdx.x == lane
    const int half = lane >> 4;       // K-pair selector (ISA f32 A/B layout)
    const int r    = lane & 15;       // M row (A) / N col (B)
    const int n0   = threadIdx.y * 16;

    const float* arow = &aTile[r * LDSROW + 2 * half];
    const float* brow = W + (size_t)(n0 + r) * FD + 2 * half;   // B[k][n] = W[n][k]

    v8f c = {};
#pragma unroll 8
    for (int k = 0; k < FD; k += 4) {
        v2f a = *(const v2f*)(arow + k);   // lanes 0-15: K=k,k+1 ; lanes 16-31: K=k+2,k+3
        v2f b = *(const v2f*)(brow + k);
        c = __builtin_amdgcn_wmma_f32_16x16x4_f32(
                /*neg_a=*/false, a, /*neg_b=*/false, b,
                /*c_mod=*/(short)0, c, /*reuse_a=*/false, /*reuse_b=*/false);
    }

    float bv = bias[n0 + r];
    float* orow = out + (size_t)(m0 + 8 * half) * NOUT + (n0 + r);
#pragma unroll
    for (int i = 0; i < 8; ++i) {
        float v = c[i] + bv;                  // VGPR i: lanes0-15 -> M=i, lanes16-31 -> M=i+8
        if (RELU) v = fmaxf(v, 0.0f);
        orow[(size_t)i * NOUT] = v;
    }
}

extern "C" void kernel_launch(void* const* d_in, const int* in_sizes, int n_in,
                              void* d_out, int out_size, void* d_ws, size_t ws_size,
                              hipStream_t stream) {
    const float* in_feat = (const float*)d_in[0];
    const int*   src     = (const int*)d_in[1];
    const int*   dst     = (const int*)d_in[2];
    const float* W0 = (const float*)d_in[3]; const float* b0 = (const float*)d_in[4];
    const float* W1 = (const float*)d_in[5]; const float* b1 = (const float*)d_in[6];
    const float* W2 = (const float*)d_in[7]; const float* b2 = (const float*)d_in[8];
    float* out = (float*)d_out;

    // workspace layout (all offsets 16B-aligned)
    float* deg   = (float*)d_ws;                  // NN floats
    float* neigh = deg + NN;                      // NN*FD
    float* xbuf  = neigh + (size_t)NN * FD;       // NN*FD (layer0/1 outputs ping here)

    const int T = 256;
    // degree (shared across all three layers)
    k_init_deg <<<(NN + T - 1) / T, T, 0, stream>>>(deg);
    k_count_deg<<<(NE + T - 1) / T, T, 0, stream>>>(dst, deg);

    const int agg_blocks = (NE * 32) / T;       // 100000
    const int vec_blocks = (NN * FD / 4) / T;   // 6250
    const int m_tiles    = NN / 16;             // 3125 exactly

    // ---- layer 0: x = in_feat -> xbuf (relu, 128 out) ----
    k_init_neigh<<<vec_blocks, T, 0, stream>>>(in_feat, neigh);
    k_aggregate <<<agg_blocks, T, 0, stream>>>(in_feat, src, dst, neigh);
    k_gemm_wmma<128, true><<<m_tiles, dim3(32, 8), 0, stream>>>(neigh, deg, W0, b0, xbuf);

    // ---- layer 1: x = xbuf -> xbuf (relu, 128 out) ----
    k_init_neigh<<<vec_blocks, T, 0, stream>>>(xbuf, neigh);
    k_aggregate <<<agg_blocks, T, 0, stream>>>(xbuf, src, dst, neigh);
    k_gemm_wmma<128, true><<<m_tiles, dim3(32, 8), 0, stream>>>(neigh, deg, W1, b1, xbuf);

    // ---- layer 2: x = xbuf -> d_out (no relu, 64 out) ----
    k_init_neigh<<<vec_blocks, T, 0, stream>>>(xbuf, neigh);
    k_aggregate <<<agg_blocks, T, 0, stream>>>(xbuf, src, dst, neigh);
    k_gemm_wmma<64, false><<<m_tiles, dim3(32, 4), 0, stream>>>(neigh, deg, W2, b2, out);
}